// MultiHeadAttention_45071386804487
// MI455X (gfx1250) — compile-verified
//
#include <hip/hip_runtime.h>
#include <math.h>
#include <stdint.h>

typedef __attribute__((ext_vector_type(2))) float v2f;
typedef __attribute__((ext_vector_type(8))) float v8f;
typedef __attribute__((ext_vector_type(4))) unsigned int u32x4;
typedef __attribute__((ext_vector_type(8))) int i32x8;

#define WMMA_F32(a, b, c) \
  __builtin_amdgcn_wmma_f32_16x16x4_f32(false, (a), false, (b), (short)0, (c), false, false)

static constexpr int BSZ   = 4;
static constexpr int T     = 2048;
static constexpr int CDIM  = 1024;
static constexpr int NH    = 16;
static constexpr int HD    = 64;          // CDIM / NH
static constexpr int LDQKV = 3 * CDIM;    // 3072

// ---------------------------------------------------------------------------
// CDNA5 async global->LDS (ASYNCcnt) and TDM (TENSORcnt) helpers.
// LDS byte address = low 32 bits of the generic pointer (flat->LDS mapping
// truncates to the wave-relative LDS offset).
// ---------------------------------------------------------------------------
__device__ __forceinline__ uint32_t lds_off(const void* p) {
  return (uint32_t)(uintptr_t)p;
}
__device__ __forceinline__ void async_b128(uint32_t dst_lds, const float* src) {
  asm volatile("global_load_async_to_lds_b128 %0, %1, off"
               :: "v"(dst_lds), "v"(src) : "memory");
}
__device__ __forceinline__ void wait_async0() {
  asm volatile("s_wait_asynccnt 0x0" ::: "memory");
}
__device__ __forceinline__ void wait_tensor0() {
#if __has_builtin(__builtin_amdgcn_s_wait_tensorcnt)
  __builtin_amdgcn_s_wait_tensorcnt(0);
#else
  asm volatile("s_wait_tensorcnt 0x0" ::: "memory");
#endif
}
// 2D tensor-DMA load: D# groups 0 and 1 in SGPR tuples (VADDR2/3 NULL -> 2D).
__device__ __forceinline__ void tdm_load_2d(u32x4 g0, i32x8 g1) {
  asm volatile("tensor_load_to_lds %0, %1" :: "s"(g0), "s"(g1) : "memory");
}
// Build D# group1 for a 2D tile: 64x16 fp32 rows with row stride = N elements.
__device__ __forceinline__ i32x8 make_tdm_g1(int N) {
  i32x8 g1;
  g1[0] = 0x20000;                              // data_size = 2 (4 bytes)
  g1[1] = (N & 0xFFFF) << 16;                   // tensor_dim0[15:0]   @ bits 63:48
  g1[2] = ((unsigned)N >> 16) | (16 << 16);     // tensor_dim0[31:16], tensor_dim1=16
  g1[3] = (64 << 16);                           // tile_dim0 = 64      @ bits 127:112
  g1[4] = 16;                                   // tile_dim1 = 16      @ bits 143:128
  g1[5] = N;                                    // tensor_dim0_stride[31:0]
  g1[6] = 0;
  g1[7] = 0;
  return g1;
}
__device__ __forceinline__ u32x4 make_tdm_g0(uint32_t lds_addr, const float* gptr) {
  const uint64_t ga = (uint64_t)(uintptr_t)gptr;
  u32x4 g0;
  g0[0] = 1u;                                   // count = 1 valid descriptor
  g0[1] = lds_addr;                             // LDS byte address
  g0[2] = (uint32_t)ga;                         // global_addr[31:0]
  g0[3] = (uint32_t)(ga >> 32) | 0x80000000u;   // global_addr[56:32] | type=2<<30
  return g0;
}

// ---------------------------------------------------------------------------
// fp32 WMMA GEMM: Out[M x N] = A[M x 1024] * W[1024 x N] + bias[N]
// Block = 256 threads = 8 waves -> 64x64 C tile (each wave: 32x16 via two
// 16x16 accumulators sharing a B fragment). K chunk = 16, double-buffered:
//   A tile (64x16): per-thread async b128 copies     (ASYNCcnt)
//   B tile (16x64): one TDM descriptor, wave 0 only  (TENSORcnt)
// ---------------------------------------------------------------------------
__global__ __launch_bounds__(256) void gemm_f32_wmma(
    const float* __restrict__ A, const float* __restrict__ W,
    const float* __restrict__ bias, float* __restrict__ Out,
    int N, int nTiles /* = N/64 */)
{
  __shared__ __attribute__((aligned(16))) float asA[2][64 * 16];
  __shared__ __attribute__((aligned(16))) float asB[2][16 * 64];

  const int tid  = threadIdx.x;
  const int lane = tid & 31;
  const int r    = lane & 15;
  const int hi   = lane >> 4;
  const int wid  = tid >> 5;
  const int msub = (wid >> 2) * 32;
  const int nsub = (wid & 3) * 16;
  const int m0   = (blockIdx.x / nTiles) * 64;
  const int n0   = (blockIdx.x % nTiles) * 64;

  // A cooperative copy map: one b128 per thread per chunk (64 rows x 16 floats).
  const int arow = tid >> 2, apart = (tid & 3) * 4;
  const float* aSrc = A + (size_t)(m0 + arow) * CDIM + apart;

  // B TDM descriptor pieces.
  const float*   bTile0 = W + n0;                     // chunk c tile = bTile0 + c*16*N
  const i32x8    g1     = make_tdm_g1(N);
  const uint32_t ldsB[2] = { lds_off(&asB[0][0]), lds_off(&asB[1][0]) };

  // Stage chunk 0.
  async_b128(lds_off(&asA[0][arow * 16 + apart]), aSrc);
  if (wid == 0) tdm_load_2d(make_tdm_g0(ldsB[0], bTile0), g1);

  v8f acc0 = {}, acc1 = {};
  constexpr int NC = CDIM / 16;   // 64 chunks
  for (int c = 0; c < NC; ++c) {
    wait_async0();
    if (wid == 0) wait_tensor0();
    __syncthreads();
    if (c + 1 < NC) {
      const int nb = (c + 1) & 1;
      async_b128(lds_off(&asA[nb][arow * 16 + apart]), aSrc + (c + 1) * 16);
      if (wid == 0)
        tdm_load_2d(make_tdm_g0(ldsB[nb], bTile0 + (size_t)(c + 1) * 16 * N), g1);
    }
    const float* at = &asA[c & 1][0];
    const float* bt = &asB[c & 1][0];
#pragma unroll
    for (int kk = 0; kk < 4; ++kk) {
      const int kd = kk * 4 + 2 * hi;
      float2 a2 = *(const float2*)(at + (msub + r) * 16 + kd);
      float2 a3 = *(const float2*)(at + (msub + 16 + r) * 16 + kd);
      v2f a0 = { a2.x, a2.y };
      v2f a1 = { a3.x, a3.y };
      v2f bb = { bt[kd * 64 + nsub + r], bt[(kd + 1) * 64 + nsub + r] };
      acc0 = WMMA_F32(a0, bb, acc0);
      acc1 = WMMA_F32(a1, bb, acc1);
    }
  }

  const float bv = bias[n0 + nsub + r];
  float* c0 = Out + (size_t)(m0 + msub + 8 * hi) * N + n0 + nsub + r;
  float* c1 = Out + (size_t)(m0 + msub + 16 + 8 * hi) * N + n0 + nsub + r;
#pragma unroll
  for (int vg = 0; vg < 8; ++vg) {
    c0[(size_t)vg * N] = acc0[vg] + bv;
    c1[(size_t)vg * N] = acc1[vg] + bv;
  }
}

// ---------------------------------------------------------------------------
// In-place RoPE on the Q and K thirds of the QKV buffer.
// ---------------------------------------------------------------------------
__global__ __launch_bounds__(256) void rope_kernel(float* __restrict__ qkv)
{
  const int idx = blockIdx.x * blockDim.x + threadIdx.x;
  const int d  = idx & 31;
  const int h  = (idx >> 5) & (NH - 1);
  const int bt = idx >> 9;
  const int t  = bt & (T - 1);

  const float freq = __expf(-(float)d * (9.210340371976184f / 32.0f)); // 10000^(-d/32)
  const float ang  = (float)t * freq;
  const float sn = sinf(ang);
  const float cs = cosf(ang);

  float* qp = qkv + (size_t)bt * LDQKV + h * HD + d;
  float x1 = qp[0], x2 = qp[32];
  qp[0]  = x1 * cs - x2 * sn;
  qp[32] = x2 * cs + x1 * sn;

  float* kp = qp + CDIM;
  x1 = kp[0]; x2 = kp[32];
  kp[0]  = x1 * cs - x2 * sn;
  kp[32] = x2 * cs + x1 * sn;
}

// ---------------------------------------------------------------------------
// Flash attention. Block = 8 waves = 8 consecutive q-tiles of one (batch,head):
// the 16x64 K and V tiles are async-staged to LDS ONCE per block (8x reuse),
// double-buffered on ASYNCcnt. Online softmax with 16-lane shfl reductions;
// P re-laid out C->A through a wave-private LDS tile.
// ---------------------------------------------------------------------------
__global__ __launch_bounds__(256) void attn_kernel(
    const float* __restrict__ qkv, float* __restrict__ y)
{
  __shared__ __attribute__((aligned(16))) float kbuf[2][16 * 64];
  __shared__ __attribute__((aligned(16))) float vbuf[2][16 * 64];
  __shared__ __attribute__((aligned(16))) float plds[8][256];

  const int tid  = threadIdx.x;
  const int lane = tid & 31;
  const int wib  = tid >> 5;
  const int qg = blockIdx.x & 15;            // q-tile group (8 tiles each)
  const int bh = blockIdx.x >> 4;            // 0..63
  const int h  = bh & (NH - 1);
  const int b  = bh >> 4;
  const int q0    = (qg * 8 + wib) * 16;     // this wave's q-tile
  const int q0max = (qg * 8 + 7) * 16;       // block's last q-tile
  const int r  = lane & 15;
  const int hi = lane >> 4;

  const float* qb  = qkv + ((size_t)b * T + q0) * LDQKV + h * HD;
  const float* kb  = qkv + (size_t)b * T * LDQKV + CDIM + h * HD;
  const float* vbp = qkv + (size_t)b * T * LDQKV + 2 * CDIM + h * HD;

  // Cooperative K/V staging map: one b128 per thread per matrix per tile.
  const int srow = tid >> 4, sseg = (tid & 15) * 4;     // 16 rows x 64 floats
  const float* kSrc = kb  + (size_t)srow * LDQKV + sseg;
  const float* vSrc = vbp + (size_t)srow * LDQKV + sseg;

  // Preload Q fragments (pre-scaled by 1/sqrt(HD) = 0.125).
  v2f qf[16];
  {
    const float* qrow = qb + (size_t)r * LDQKV + 2 * hi;
#pragma unroll
    for (int kk = 0; kk < 16; ++kk) {
      float2 t2 = *(const float2*)(qrow + kk * 4);
      v2f a = { t2.x * 0.125f, t2.y * 0.125f };
      qf[kk] = a;
    }
  }

  v8f o0 = {}, o1 = {}, o2 = {}, o3 = {};
  float mrow[8], lrow[8];
#pragma unroll
  for (int vg = 0; vg < 8; ++vg) { mrow[vg] = -__builtin_inff(); lrow[vg] = 0.f; }

  float* pl = &plds[wib][0];

  // Stage tile j0 = 0.
  async_b128(lds_off(&kbuf[0][srow * 64 + sseg]), kSrc);
  async_b128(lds_off(&vbuf[0][srow * 64 + sseg]), vSrc);

  for (int j0 = 0; j0 <= q0max; j0 += 16) {
    wait_async0();
    __syncthreads();
    if (j0 + 16 <= q0max) {
      const int nb = ((j0 >> 4) + 1) & 1;
      async_b128(lds_off(&kbuf[nb][srow * 64 + sseg]), kSrc + (size_t)(j0 + 16) * LDQKV);
      async_b128(lds_off(&vbuf[nb][srow * 64 + sseg]), vSrc + (size_t)(j0 + 16) * LDQKV);
    }
    if (j0 <= q0) {                       // wave-uniform; EXEC stays full
      const int cb = (j0 >> 4) & 1;
      const float* kt = &kbuf[cb][0];
      const float* vt = &vbuf[cb][0];

      // S = Q * K^T from LDS.
      v8f s = {};
      const float* krow = kt + r * 64 + 2 * hi;
#pragma unroll
      for (int kk = 0; kk < 16; ++kk) {
        float2 k2 = *(const float2*)(krow + kk * 4);
        v2f bb = { k2.x, k2.y };
        s = WMMA_F32(qf[kk], bb, s);
      }

      if (j0 == q0) {                     // diagonal tile: causal mask (selects)
#pragma unroll
        for (int vg = 0; vg < 8; ++vg)
          s[vg] = (r > vg + 8 * hi) ? -__builtin_inff() : s[vg];
      }

      float al[8];
#pragma unroll
      for (int vg = 0; vg < 8; ++vg) {
        float tm = s[vg];
        tm = fmaxf(tm, __shfl_xor(tm, 1, 32));
        tm = fmaxf(tm, __shfl_xor(tm, 2, 32));
        tm = fmaxf(tm, __shfl_xor(tm, 4, 32));
        tm = fmaxf(tm, __shfl_xor(tm, 8, 32));
        const float mn = fmaxf(mrow[vg], tm);
        const float a  = __expf(mrow[vg] - mn);
        const float p  = __expf(s[vg] - mn);
        float rs = p;
        rs += __shfl_xor(rs, 1, 32);
        rs += __shfl_xor(rs, 2, 32);
        rs += __shfl_xor(rs, 4, 32);
        rs += __shfl_xor(rs, 8, 32);
        lrow[vg] = lrow[vg] * a + rs;
        mrow[vg] = mn;
        al[vg]   = a;
        pl[(vg + 8 * hi) * 16 + r] = p;   // stage P in C layout
      }

#pragma unroll
      for (int vg = 0; vg < 8; ++vg) {
        o0[vg] *= al[vg]; o1[vg] *= al[vg]; o2[vg] *= al[vg]; o3[vg] *= al[vg];
      }

      // Reload P in A layout (per-wave LDS ops are in-order).
      v2f pa[4];
      {
        const float* prow = pl + r * 16 + 2 * hi;
#pragma unroll
        for (int kk = 0; kk < 4; ++kk) {
          v2f a = { prow[kk * 4], prow[kk * 4 + 1] };
          pa[kk] = a;
        }
      }

      // O += P * V from LDS.
#pragma unroll
      for (int kk = 0; kk < 4; ++kk) {
        const float* vr = vt + (kk * 4 + 2 * hi) * 64 + r;
        v2f b0 = { vr[0],  vr[64] };
        v2f b1 = { vr[16], vr[80] };
        v2f b2 = { vr[32], vr[96] };
        v2f b3 = { vr[48], vr[112] };
        o0 = WMMA_F32(pa[kk], b0, o0);
        o1 = WMMA_F32(pa[kk], b1, o1);
        o2 = WMMA_F32(pa[kk], b2, o2);
        o3 = WMMA_F32(pa[kk], b3, o3);
      }
    }
  }

#pragma unroll
  for (int vg = 0; vg < 8; ++vg) {
    const float inv = 1.0f / lrow[vg];
    float* yr = y + ((size_t)b * T + q0 + vg + 8 * hi) * CDIM + h * HD + r;
    yr[0]  = o0[vg] * inv;
    yr[16] = o1[vg] * inv;
    yr[32] = o2[vg] * inv;
    yr[48] = o3[vg] * inv;
  }
}

// ---------------------------------------------------------------------------
extern "C" void kernel_launch(void* const* d_in, const int* in_sizes, int n_in,
                              void* d_out, int out_size, void* d_ws, size_t ws_size,
                              hipStream_t stream)
{
  const float* x     = (const float*)d_in[0];
  const float* Wqkv  = (const float*)d_in[1];
  const float* bqkv  = (const float*)d_in[2];
  const float* Wproj = (const float*)d_in[3];
  const float* bproj = (const float*)d_in[4];
  float* out = (float*)d_out;

  float* qkv = (float*)d_ws;                          // 8192 x 3072 fp32 (96 MB)
  float* yws = qkv + (size_t)BSZ * T * LDQKV;         // 8192 x 1024 fp32 (32 MB)

  const int M = BSZ * T;                              // 8192
  // QKV GEMM: (8192/64) x (3072/64) = 128 x 48 = 6144 blocks
  gemm_f32_wmma<<<(M / 64) * (LDQKV / 64), 256, 0, stream>>>(x, Wqkv, bqkv, qkv, LDQKV, LDQKV / 64);
  // RoPE on Q,K thirds: B*T*NH*32 threads
  rope_kernel<<<(BSZ * T * NH * 32) / 256, 256, 0, stream>>>(qkv);
  // Flash attention: 64 (b,h) x 16 q-groups = 1024 blocks
  attn_kernel<<<BSZ * NH * (T / 128), 256, 0, stream>>>(qkv, yws);
  // Output projection: 128 x 16 = 2048 blocks
  gemm_f32_wmma<<<(M / 64) * (CDIM / 64), 256, 0, stream>>>(yws, Wproj, bproj, out, CDIM, CDIM / 64);
}